// GCN_4887672783345
// MI455X (gfx1250) — compile-verified
//
#include <hip/hip_runtime.h>

#define D 64
#define DOUT 16
#define WPITCH 72   // LDS row pitch (floats): 2*72 mod 64 = 16 -> half-waves hit disjoint bank sets

typedef __attribute__((ext_vector_type(2))) float v2f;
typedef __attribute__((ext_vector_type(4))) float v4f;
typedef __attribute__((ext_vector_type(8))) float v8f;

// ---------------- degree / normalization ----------------
__global__ void deg_init(float* __restrict__ deg, int n) {
    int i = blockIdx.x * blockDim.x + threadIdx.x;
    if (i < n) deg[i] = 1.0f;            // self-loop contributes 1
}

__global__ void deg_accum(const long long* __restrict__ dst,
                          float* __restrict__ deg, int e) {
    int i = blockIdx.x * blockDim.x + threadIdx.x;
    if (i < e) atomicAdd(&deg[(int)dst[i]], 1.0f);
}

__global__ void deg_rsqrt(float* __restrict__ deg, int n) {
    int i = blockIdx.x * blockDim.x + threadIdx.x;
    if (i < n) deg[i] = rsqrtf(deg[i]);  // deg >= 1 always (self-loop)
}

// ---------------- X[n x 64] @ W[64 x 64] via f32 WMMA, W staged in LDS ----
// One wave computes a 16-row x 64-col tile. relu!=0 applies ReLU to the
// A operand on load (fuses previous layer's activation).
__global__ void gemm64(const float* __restrict__ A, const float* __restrict__ W,
                       float* __restrict__ C, int n, int relu) {
    __shared__ float Wl[D * WPITCH];     // 18 KB of the 320 KB WGP LDS

    // Cooperative stage of W (64x64 f32 = 1024 float4), repitched to WPITCH.
    for (int i = threadIdx.x; i < D * D / 4; i += blockDim.x) {
        int row  = i >> 4;               // /16 float4s per row
        int col4 = i & 15;
        *(v4f*)(Wl + row * WPITCH + col4 * 4) = ((const v4f*)W)[i];
    }
    __syncthreads();

    const int wave    = (blockIdx.x * blockDim.x + threadIdx.x) >> 5;
    const int lane    = threadIdx.x & 31;
    const int rowBase = wave << 4;
    if (rowBase >= n) return;            // wave-uniform (after barrier): EXEC all-1 around WMMA
    const int half = lane >> 4;          // 0: K={k0,k0+1}, 1: K={k0+2,k0+3}
    const int lr   = lane & 15;

    v8f acc0 = {}, acc1 = {}, acc2 = {}, acc3 = {};
    const float* arow = A + (size_t)(rowBase + lr) * D;

    for (int ks = 0; ks < 16; ++ks) {
        const int k0 = ks * 4 + half * 2;
        float a0 = arow[k0], a1 = arow[k0 + 1];
        if (relu) { a0 = fmaxf(a0, 0.0f); a1 = fmaxf(a1, 0.0f); }
        v2f a = { a0, a1 };
        const float* w0 = Wl + k0 * WPITCH + lr;    // row k0 of W (LDS)
        const float* w1 = w0 + WPITCH;              // row k0+1
        v2f b0 = { w0[0],  w1[0]  };
        v2f b1 = { w0[16], w1[16] };
        v2f b2 = { w0[32], w1[32] };
        v2f b3 = { w0[48], w1[48] };
        acc0 = __builtin_amdgcn_wmma_f32_16x16x4_f32(false, a, false, b0, (short)0, acc0, false, false);
        acc1 = __builtin_amdgcn_wmma_f32_16x16x4_f32(false, a, false, b1, (short)0, acc1, false, false);
        acc2 = __builtin_amdgcn_wmma_f32_16x16x4_f32(false, a, false, b2, (short)0, acc2, false, false);
        acc3 = __builtin_amdgcn_wmma_f32_16x16x4_f32(false, a, false, b3, (short)0, acc3, false, false);
    }
    // C/D layout: VGPR v holds row (v + 8*half), col lr of the 16x16 tile
    for (int v = 0; v < 8; ++v) {
        float* crow = C + (size_t)(rowBase + v + 8 * half) * D + lr;
        crow[0]  = acc0[v];
        crow[16] = acc1[v];
        crow[32] = acc2[v];
        crow[48] = acc3[v];
    }
}

// ---------------- head: H[n x 64] @ Wh[64 x 16] + bh ----------------
// Wh staged in LDS with natural pitch 16 (row k0 vs k0+2 -> bank shift 32,
// the two half-waves already use disjoint bank sets).
__global__ void head16(const float* __restrict__ A, const float* __restrict__ Wh,
                       const float* __restrict__ bh, float* __restrict__ C, int n) {
    __shared__ float Wl[D * DOUT];
    for (int i = threadIdx.x; i < D * DOUT / 4; i += blockDim.x)
        ((v4f*)Wl)[i] = ((const v4f*)Wh)[i];
    __syncthreads();

    const int wave    = (blockIdx.x * blockDim.x + threadIdx.x) >> 5;
    const int lane    = threadIdx.x & 31;
    const int rowBase = wave << 4;
    if (rowBase >= n) return;
    const int half = lane >> 4;
    const int lr   = lane & 15;

    v8f acc = {};
    const float* arow = A + (size_t)(rowBase + lr) * D;
    for (int ks = 0; ks < 16; ++ks) {
        const int k0 = ks * 4 + half * 2;
        v2f a = { arow[k0], arow[k0 + 1] };
        v2f b = { Wl[k0 * DOUT + lr], Wl[(k0 + 1) * DOUT + lr] };
        acc = __builtin_amdgcn_wmma_f32_16x16x4_f32(false, a, false, b, (short)0, acc, false, false);
    }
    const float bias = bh[lr];
    for (int v = 0; v < 8; ++v)
        C[(size_t)(rowBase + v + 8 * half) * DOUT + lr] = acc[v] + bias;
}

// ---------------- aggregation init: bias + self-loop term ----------------
__global__ void agg_init(const float* __restrict__ tmp, const float* __restrict__ dinv,
                         const float* __restrict__ bias, float* __restrict__ H, int n) {
    int idx = blockIdx.x * blockDim.x + threadIdx.x;
    if (idx >= n * D) return;
    int node = idx >> 6;
    int f    = idx & (D - 1);
    float di = dinv[node];
    H[idx] = bias[f] + di * di * tmp[idx];
}

// ---------------- edge scatter: H[dst] += dinv[s]*dinv[d]*tmp[src] ----------------
// One wave per edge; lane handles features lane and lane+32 (coalesced).
__global__ void scatter(const long long* __restrict__ srcIdx,
                        const long long* __restrict__ dstIdx,
                        const float* __restrict__ dinv,
                        const float* __restrict__ tmp,
                        float* __restrict__ H, int e) {
    int t    = blockIdx.x * blockDim.x + threadIdx.x;
    int edge = t >> 5;
    int lane = t & 31;
    if (edge >= e) return;
    int s = (int)srcIdx[edge];
    int d = (int)dstIdx[edge];
    float nrm = dinv[s] * dinv[d];
    const float* m = tmp + (size_t)s * D;
    float*       o = H   + (size_t)d * D;
    atomicAdd(o + lane,      nrm * m[lane]);
    atomicAdd(o + lane + 32, nrm * m[lane + 32]);
}

extern "C" void kernel_launch(void* const* d_in, const int* in_sizes, int n_in,
                              void* d_out, int out_size, void* d_ws, size_t ws_size,
                              hipStream_t stream) {
    const float*     x   = (const float*)d_in[0];
    const long long* ei  = (const long long*)d_in[1];   // int64 [2, E]
    const float*     W1  = (const float*)d_in[2];
    const float*     b1  = (const float*)d_in[3];
    const float*     W2  = (const float*)d_in[4];
    const float*     b2  = (const float*)d_in[5];
    const float*     W3  = (const float*)d_in[6];
    const float*     b3  = (const float*)d_in[7];
    const float*     Wh  = (const float*)d_in[8];
    const float*     bh  = (const float*)d_in[9];
    float*           out = (float*)d_out;

    const int n = in_sizes[0] / D;        // 50000
    const int e = in_sizes[1] / 2;        // 800000
    const long long* srcIdx = ei;         // edge_index[0]
    const long long* dstIdx = ei + e;     // edge_index[1]

    // workspace layout (256B aligned): dinv[n] | tmp[n*64] | H[n*64]
    char*  ws   = (char*)d_ws;
    size_t o0   = 0;
    size_t o1   = (o0 + (size_t)n * 4 + 255) & ~(size_t)255;
    size_t o2   = (o1 + (size_t)n * D * 4 + 255) & ~(size_t)255;
    float* dinv = (float*)(ws + o0);
    float* tmp  = (float*)(ws + o1);
    float* H    = (float*)(ws + o2);

    const int TB = 256;
    const int nBlk  = (n + TB - 1) / TB;
    const int eBlk  = (e + TB - 1) / TB;
    const int ntBlk = (n * D + TB - 1) / TB;
    const int gemmWaves = (n + 15) / 16;
    const int gemmBlk   = (gemmWaves * 32 + TB - 1) / TB;
    const int scatBlk   = (int)(((long long)e * 32 + TB - 1) / TB);

    // normalization: dinv = rsqrt(1 + indegree)
    deg_init <<<nBlk, TB, 0, stream>>>(dinv, n);
    deg_accum<<<eBlk, TB, 0, stream>>>(dstIdx, dinv, e);
    deg_rsqrt<<<nBlk, TB, 0, stream>>>(dinv, n);

    // layer 1
    gemm64  <<<gemmBlk, TB, 0, stream>>>(x, W1, tmp, n, 0);
    agg_init<<<ntBlk,   TB, 0, stream>>>(tmp, dinv, b1, H, n);
    scatter <<<scatBlk, TB, 0, stream>>>(srcIdx, dstIdx, dinv, tmp, H, e);
    // layer 2 (ReLU of layer-1 output fused into GEMM A-load)
    gemm64  <<<gemmBlk, TB, 0, stream>>>(H, W2, tmp, n, 1);
    agg_init<<<ntBlk,   TB, 0, stream>>>(tmp, dinv, b2, H, n);
    scatter <<<scatBlk, TB, 0, stream>>>(srcIdx, dstIdx, dinv, tmp, H, e);
    // layer 3
    gemm64  <<<gemmBlk, TB, 0, stream>>>(H, W3, tmp, n, 1);
    agg_init<<<ntBlk,   TB, 0, stream>>>(tmp, dinv, b3, H, n);
    scatter <<<scatBlk, TB, 0, stream>>>(srcIdx, dstIdx, dinv, tmp, H, e);
    // head (no ReLU after layer 3)
    head16  <<<gemmBlk, TB, 0, stream>>>(H, Wh, bh, out, n);
}